// ChamferLoss_67740224192622
// MI455X (gfx1250) — compile-verified
//
#include <hip/hip_runtime.h>
#include <hip/hip_bf16.h>

// Chamfer loss for B=16 batches of N=4096 3-D points.
// Distance tile via V_WMMA_F32_16X16X4_F32 using the K=4 embedding:
//   d(n,m) = [-2x0,-2x1,-2x2,1]_n . [y0,y1,y2,|y|^2]_m + |x_n|^2 (as C).

typedef float v2f __attribute__((ext_vector_type(2)));
typedef float v8f __attribute__((ext_vector_type(8)));

#define ROW_TILES_PER_WAVE 4   // 4 row-tiles (64 rows) per wave
#define FLT_BIG 3.402823466e+38f

// Raw v_min_num_f32: avoids the v_max_num_f32(x,x) canonicalize pair that
// fminf's IEEE-minNum semantics force.  Inputs here are never sNaN (WMMA
// outputs on finite data, or FLT_MAX init), so this is safe.
__device__ __forceinline__ float minf32(float a, float b) {
    asm("v_min_num_f32 %0, %0, %1" : "+v"(a) : "v"(b));
    return a;
}

__global__ void chamfer_zero_kernel(float* out) { out[0] = 0.0f; }

// Precompute |p|^2 for every point of a [B,3,N] cloud into sq[B*N].
__global__ void chamfer_sqnorm_kernel(const float* __restrict__ P,
                                      float* __restrict__ sq, int N)
{
    const int i = blockIdx.x * blockDim.x + threadIdx.x;   // 0 .. B*N-1
    const int batch = i / N;
    const int n     = i - batch * N;
    const float* p0 = P + (size_t)batch * 3 * N;
    const float c0 = p0[n], c1 = p0[N + n], c2 = p0[2 * N + n];
    sq[i] = c0 * c0 + c1 * c1 + c2 * c2;
}

// Computes  scale * sum_n min_m ||x_n - y_m||^2  per batch, accumulated into
// out[0].  The symmetric chamfer term is the same kernel with X/Y swapped.
// USE_YY: |y|^2 was precomputed into Ysq (2 loads/tile, no VALU for the B
// fragment); otherwise it is computed inline (4 loads + 3 VALU + 1 select).
template <bool USE_YY>
__global__ __launch_bounds__(32)
void chamfer_dirmin_kernel(const float* __restrict__ X,
                           const float* __restrict__ Y,
                           const float* __restrict__ Ysq,
                           float* __restrict__ out,
                           int N, float scale)
{
    const int lane = threadIdx.x;          // 0..31 (wave32, one wave per block)
    const int l    = lane & 15;
    const bool hi  = lane >= 16;

    const int groupsPerBatch = N / (16 * ROW_TILES_PER_WAVE);
    const int batch  = blockIdx.x / groupsPerBatch;
    const int group  = blockIdx.x % groupsPerBatch;
    const int nbase0 = group * 16 * ROW_TILES_PER_WAVE;

    const float* x0 = X + (size_t)batch * 3 * N;
    const float* x1 = x0 + N;
    const float* x2 = x1 + N;
    const float* y0 = Y + (size_t)batch * 3 * N;
    const float* y1 = y0 + N;
    const float* y2 = y1 + N;
    const float* yy = Ysq + (size_t)batch * N;

    // ---- Build A fragments and C fragments for the 4 row tiles -------------
    // A 16x4 f32 layout: lanes 0-15: M=lane, K=0 (v.x) / K=1 (v.y)
    //                    lanes16-31: M=lane-16, K=2 (v.x) / K=3 (v.y)
    // Row n of A = (-2*x0, -2*x1, -2*x2, 1).
    // C/D layout: VGPR v, lanes 0-15 -> row v, lanes 16-31 -> row v+8.
    v2f afrag[ROW_TILES_PER_WAVE];
    v8f cfrag[ROW_TILES_PER_WAVE];
    float rowmin[ROW_TILES_PER_WAVE][8];

    #pragma unroll
    for (int r = 0; r < ROW_TILES_PER_WAVE; ++r) {
        const int nb = nbase0 + r * 16;
        const float ax0 = x0[nb + l];
        const float ax1 = x1[nb + l];
        const float ax2 = x2[nb + l];
        const float xxl = ax0 * ax0 + ax1 * ax1 + ax2 * ax2; // |x_row|^2, row = l

        afrag[r].x = hi ? (-2.0f * ax2) : (-2.0f * ax0);
        afrag[r].y = hi ? 1.0f          : (-2.0f * ax1);

        #pragma unroll
        for (int v = 0; v < 8; ++v) {
            const int row = v + (hi ? 8 : 0);
            cfrag[r][v] = __shfl(xxl, row, 32);   // broadcast |x_row|^2 across columns
        }
        #pragma unroll
        for (int v = 0; v < 8; ++v) rowmin[r][v] = FLT_BIG;
    }

    // Lane-half-dependent B-fragment sources, selected once:
    // B 4x16 f32 layout: V0 = { K=0 row | K=2 row }, V1 = { K=1 row | K=3 row }
    // Column m of B = (y0, y1, y2, |y|^2).
    const float* pbx = hi ? y2 : y0;          // V0: y0 (lo lanes) / y2 (hi lanes)
    const float* pby = hi ? yy : y1;          // V1: y1 (lo lanes) / |y|^2 (hi lanes)

    // ---- Sweep all column tiles; one B-fragment load feeds 4 WMMAs ---------
    for (int mbase = 0; mbase < N; mbase += 16) {
        v2f bfrag;
        if (USE_YY) {
            bfrag.x = pbx[mbase + l];
            bfrag.y = pby[mbase + l];
        } else {
            const float by0 = y0[mbase + l];
            const float by1 = y1[mbase + l];
            const float by2 = y2[mbase + l];
            const float byy = by0 * by0 + by1 * by1 + by2 * by2;
            bfrag.x = pbx[mbase + l];
            bfrag.y = hi ? byy : by1;
        }

        #pragma unroll
        for (int r = 0; r < ROW_TILES_PER_WAVE; ++r) {
            // D = A x B + C  -> full 16x16 tile of squared distances
            v8f d = __builtin_amdgcn_wmma_f32_16x16x4_f32(
                false, afrag[r], false, bfrag,
                (short)0, cfrag[r], false, false);

            // Deferred column-min: per-lane running min over tiles; the
            // cross-lane (within-tile-row) min is done once at the end.
            #pragma unroll
            for (int v = 0; v < 8; ++v)
                rowmin[r][v] = minf32(rowmin[r][v], d[v]);
        }
    }

    // ---- Final reduction: min across the 16 lanes of each half, then sum ---
    float part = 0.0f;
    #pragma unroll
    for (int r = 0; r < ROW_TILES_PER_WAVE; ++r) {
        #pragma unroll
        for (int v = 0; v < 8; ++v) {
            float t = rowmin[r][v];
            t = minf32(t, __shfl_xor(t, 1, 32));
            t = minf32(t, __shfl_xor(t, 2, 32));
            t = minf32(t, __shfl_xor(t, 4, 32));
            t = minf32(t, __shfl_xor(t, 8, 32));  // xor<16 stays within half
            part += t;   // lanes 0-15: rows 0-7 of tile; lanes 16-31: rows 8-15
        }
    }
    part += __shfl_xor(part, 16, 32);             // combine the two halves
    if (lane == 0) atomicAdd(out, part * scale);
}

extern "C" void kernel_launch(void* const* d_in, const int* in_sizes, int n_in,
                              void* d_out, int out_size, void* d_ws, size_t ws_size,
                              hipStream_t stream) {
    const float* a = (const float*)d_in[0];   // [B, 3, N] f32
    const float* b = (const float*)d_in[1];   // [B, 3, N] f32
    float* out = (float*)d_out;               // single f32 loss

    const int B = 16;
    const int N = in_sizes[0] / (B * 3);      // 4096
    const float scale = 1.0f / (float)B;
    const int grid = B * (N / (16 * ROW_TILES_PER_WAVE));  // 1024 one-wave blocks

    chamfer_zero_kernel<<<1, 1, 0, stream>>>(out);

    const size_t sqBytes = (size_t)2 * B * N * sizeof(float);   // 512 KB
    if (ws_size >= sqBytes) {
        float* sqa = (float*)d_ws;            // |a_n|^2, B*N floats
        float* sqb = sqa + (size_t)B * N;     // |b_m|^2, B*N floats
        const int total = B * N;
        chamfer_sqnorm_kernel<<<total / 256, 256, 0, stream>>>(a, sqa, N);
        chamfer_sqnorm_kernel<<<total / 256, 256, 0, stream>>>(b, sqb, N);
        // term 1: sum_n min_m d(n,m)      (Y = b, needs |b|^2)
        chamfer_dirmin_kernel<true><<<grid, 32, 0, stream>>>(a, b, sqb, out, N, scale);
        // term 2: sum_m min_n d(n,m)      (Y = a, needs |a|^2)
        chamfer_dirmin_kernel<true><<<grid, 32, 0, stream>>>(b, a, sqa, out, N, scale);
    } else {
        chamfer_dirmin_kernel<false><<<grid, 32, 0, stream>>>(a, b, nullptr, out, N, scale);
        chamfer_dirmin_kernel<false><<<grid, 32, 0, stream>>>(b, a, nullptr, out, N, scale);
    }
}